// LocalConcatSheafLearnerVariant_55422257987593
// MI455X (gfx1250) — compile-verified
//
#include <hip/hip_runtime.h>
#include <math.h>

// LocalConcatSheafLearnerVariant: per-edge gather + sum-pool + [128]x[128,9] GEMM + tanh.
// fp32 WMMA (v_wmma_f32_16x16x4_f32) keeps reference precision; kernel is L2-gather bound.

typedef __attribute__((ext_vector_type(2))) float v2f;
typedef __attribute__((ext_vector_type(8))) float v8f;

#define DH    192   // D*H
#define TWOH  128   // 2*H  (K dimension)
#define NPAD  16    // 9 outputs padded to 16 (N dimension)
#define EPW   16    // edges per wave (M dimension)
#define WAVES 4     // waves per block
#define EPB   (EPW * WAVES)   // 64 edges per block

__global__ __launch_bounds__(WAVES * 32)
void sheaf_maps_kernel(const float* __restrict__ x,
                       const float* __restrict__ W,
                       const int*   __restrict__ ei,
                       float*       __restrict__ out,
                       int E_)
{
    __shared__ float ldsW[TWOH][NPAD];            // W^T, zero-padded: 8 KB
    __shared__ float sums[WAVES][EPW][TWOH];      // pooled tiles:    32 KB

    const int tid  = threadIdx.x;
    const int lane = tid & 31;
    const int wave = tid >> 5;

    // ---- Phase 0: stage W^T (padded to 16 cols) into LDS, once per block ----
    // W is [9][128] row-major; ldsW[k][n] = (n < 9) ? W[n][k] : 0
    for (int i = tid; i < TWOH * NPAD; i += WAVES * 32) {
        const int k = i >> 4;
        const int n = i & 15;
        ldsW[k][n] = (n < 9) ? W[n * TWOH + k] : 0.0f;
    }

    const int edgeBase = blockIdx.x * EPB + wave * EPW;

    // ---- Phase 1: gather from x (L2-resident) and sum-pool into LDS ----
    // Lane owns contiguous k-slice [4*lane, 4*lane+3]. Lanes 0-15 have k<64,
    // lanes 16-31 have k>=64, so the pooling formula is lane-uniform:
    //   k <  64: sum[k] = xr[k] + xr[128+k] + xc[64+k]
    //   k >= 64: sum[k] = xr[k] + xc[k-64]  + xc[64+k]
    const int k0 = lane * 4;
    for (int m = 0; m < EPW; ++m) {
        const int e = edgeBase + m;
        float4 s = make_float4(0.f, 0.f, 0.f, 0.f);
        if (e < E_) {
            const int r = ei[e];        // edge_index[0][e]
            const int c = ei[E_ + e];   // edge_index[1][e]
            const float* xr = x + (long)r * DH;
            const float* xc = x + (long)c * DH;
            const float4 t1 = *(const float4*)(xr + k0);
            const float4 t2 = (k0 < 64) ? *(const float4*)(xr + 128 + k0)
                                        : *(const float4*)(xc + (k0 - 64));
            const float4 t3 = *(const float4*)(xc + 64 + k0);
            s.x = t1.x + t2.x + t3.x;
            s.y = t1.y + t2.y + t3.y;
            s.z = t1.z + t2.z + t3.z;
            s.w = t1.w + t2.w + t3.w;
        }
        *(float4*)&sums[wave][m][k0] = s;
    }
    __syncthreads();

    // ---- Phase 2: D[16x16] = sum[16x128] @ Wt[128x16] via 32 x wmma_f32_16x16x4_f32 ----
    // A layout (32-bit 16x4): lanes 0-15 M=lane K={0,1}; lanes 16-31 M=lane-16 K={2,3}
    // B layout (32-bit 4x16): lanes 0-15 N=lane K={0,1}; lanes 16-31 N=lane-16 K={2,3}
    v8f acc = {0.f, 0.f, 0.f, 0.f, 0.f, 0.f, 0.f, 0.f};
    const int mA = lane & 15;
    const int nB = lane & 15;
    const int kh = (lane >> 4) * 2;   // 0 or 2 within the K=4 slab

    #pragma unroll
    for (int kk = 0; kk < TWOH / 4; ++kk) {
        const int kb = kk * 4 + kh;
        v2f a, b;
        a.x = sums[wave][mA][kb];
        a.y = sums[wave][mA][kb + 1];
        b.x = ldsW[kb][nB];
        b.y = ldsW[kb + 1][nB];
        acc = __builtin_amdgcn_wmma_f32_16x16x4_f32(
            /*neg_a=*/false, a, /*neg_b=*/false, b,
            /*c_mod=*/(short)0, acc, /*reuse_a=*/false, /*reuse_b=*/false);
    }

    // ---- Phase 3: tanh + scatter store (only n < 9 valid) ----
    // C/D layout: lane holds N = lane&15; VGPR v holds M = v + 8*(lane>>4)
    const int n = lane & 15;
    if (n < 9) {
        const int mHi = 8 * (lane >> 4);
        #pragma unroll
        for (int v = 0; v < 8; ++v) {
            const int e = edgeBase + mHi + v;
            if (e < E_) {
                out[(long)e * 9 + n] = tanhf(acc[v]);
            }
        }
    }
}

extern "C" void kernel_launch(void* const* d_in, const int* in_sizes, int n_in,
                              void* d_out, int out_size, void* d_ws, size_t ws_size,
                              hipStream_t stream) {
    const float* x  = (const float*)d_in[0];   // [N, 192] f32
    const float* W  = (const float*)d_in[1];   // [9, 128] f32
    const int*   ei = (const int*)d_in[2];     // [2, E]
    float*       out = (float*)d_out;          // [E, 3, 3] f32

    const int E_ = in_sizes[2] / 2;
    const int blocks = (E_ + EPB - 1) / EPB;
    sheaf_maps_kernel<<<dim3(blocks), dim3(WAVES * 32), 0, stream>>>(x, W, ei, out, E_);
}